// DistributedGCN_4440996184260
// MI455X (gfx1250) — compile-verified
//
#include <hip/hip_runtime.h>
#include <hip/hip_bf16.h>

typedef __attribute__((ext_vector_type(2))) float v2f;
typedef __attribute__((ext_vector_type(8))) float v8f;

// ---------------------------------------------------------------------------
// WMMA GEMM + per-row scale:  H[i, n] = (sum_k X[i,k] * W[k,n]) * dis[i]
//
// Block = 128 threads (4 waves). The whole W matrix (K x N, <= 64 KB) is
// staged into LDS once per block; each wave then computes a full 16 x N
// output strip (NT = N/16 accumulator tiles) with V_WMMA_F32_16X16X4_F32,
// so every A fragment is loaded from global exactly once and reused across
// NT WMMAs, and B fragments come from LDS (ds_load).
// ---------------------------------------------------------------------------
template <int NT>   // number of 16-wide N tiles; N = NT*16
__global__ void __launch_bounds__(128)
gcn_gemm_scale_wmma(const float* __restrict__ X,
                    const float* __restrict__ W,
                    const float* __restrict__ dis,
                    float* __restrict__ H,
                    int NL, int K) {
    constexpr int N = NT * 16;
    extern __shared__ float smemW[];            // K * N floats

    // ---- cooperative W -> LDS stage (float4 vectorized) ----
    {
        const int nv4 = (K * N) >> 2;
        const float4* __restrict__ W4 = reinterpret_cast<const float4*>(W);
        float4* __restrict__ S4 = reinterpret_cast<float4*>(smemW);
        for (int i = threadIdx.x; i < nv4; i += 128) S4[i] = W4[i];
    }
    __syncthreads();

    const int lane   = threadIdx.x & 31;
    const int widblk = threadIdx.x >> 5;
    const int mTile  = blockIdx.x * 4 + widblk; // one m-tile per wave
    const int m0     = mTile << 4;
    if (m0 >= NL) return;                       // wave-uniform: EXEC all-ones

    const int half = lane >> 4;                 // 0: lanes 0-15, 1: lanes 16-31
    const int l16  = lane & 15;

    // A-fragment source row (clamped for the partial last M tile; the
    // duplicated rows are never stored).
    int rowA = m0 + l16;
    if (rowA >= NL) rowA = NL - 1;
    const float* __restrict__ xrow = X + (long long)rowA * K;

    v8f acc[NT];
#pragma unroll
    for (int t = 0; t < NT; ++t) acc[t] = v8f{};

    for (int k = 0; k < K; k += 4) {
        // A 16x4 f32 layout: VGPR0 = K{k | k+2}, VGPR1 = K{k+1 | k+3}
        const int ka = k + (half << 1);
        v2f a;
        a.x = xrow[ka + 0];
        a.y = xrow[ka + 1];
        const float* __restrict__ wr0 = smemW + ka * N + l16;
#pragma unroll
        for (int t = 0; t < NT; ++t) {
            // B 4x16 f32 layout (row striped across lanes, symmetric with A)
            v2f b;
            b.x = wr0[t * 16];
            b.y = wr0[t * 16 + N];
            acc[t] = __builtin_amdgcn_wmma_f32_16x16x4_f32(
                         /*neg_a=*/false, a, /*neg_b=*/false, b,
                         /*c_mod=*/(short)0, acc[t],
                         /*reuse_a=*/false, /*reuse_b=*/false);
        }
    }

    // D layout: VGPR r -> rows {m0+r (lanes 0-15), m0+r+8 (lanes 16-31)}
#pragma unroll
    for (int r = 0; r < 8; ++r) {
        const int row = m0 + r + (half << 3);
        if (row < NL) {
            const float d = dis[row];
            float* __restrict__ hrow = H + (long long)row * N + l16;
#pragma unroll
            for (int t = 0; t < NT; ++t) hrow[t * 16] = acc[t][r] * d;
        }
    }
}

// ---------------------------------------------------------------------------
// Zero a float buffer (agg accumulators must be re-zeroed every call).
// ---------------------------------------------------------------------------
__global__ void zero_f32(float* __restrict__ p, long long n) {
    long long i = (long long)blockIdx.x * blockDim.x + threadIdx.x;
    if (i < n) p[i] = 0.0f;
}

// ---------------------------------------------------------------------------
// Edge scatter-add: agg[row, :] += h[col, :].
// One thread = one edge x 4 channels; consecutive lanes cover consecutive
// 16B chunks of one source row -> coalesced float4 gathers, then 4
// global_atomic_add_f32. Working set (28 MB) lives in the 192 MB L2.
// ---------------------------------------------------------------------------
__global__ void scatter_add_edges(const float* __restrict__ h,
                                  const int* __restrict__ erow,
                                  const int* __restrict__ ecol,
                                  float* __restrict__ agg,
                                  int ne, int C) {
    const int groups = C >> 2;  // float4 groups per row
    long long idx = (long long)blockIdx.x * blockDim.x + threadIdx.x;
    long long total = (long long)ne * groups;
    if (idx >= total) return;
    const int e = (int)(idx / groups);
    const int g = (int)(idx - (long long)e * groups);
    const int col = ecol[e];
    const int row = erow[e];
    const float4 v = *reinterpret_cast<const float4*>(h + (long long)col * C + (g << 2));
    float* dst = agg + (long long)row * C + (g << 2);
    atomicAdd(dst + 0, v.x);
    atomicAdd(dst + 1, v.y);
    atomicAdd(dst + 2, v.z);
    atomicAdd(dst + 3, v.w);
}

// ---------------------------------------------------------------------------
// x2[i,c] = (i < owned) ? relu(agg[i,c]*dis[i] + b[c]) : 0   (halo zero-pad)
// ---------------------------------------------------------------------------
__global__ void relu_bias_pad(const float* __restrict__ agg,
                              const float* __restrict__ dis,
                              const float* __restrict__ b,
                              float* __restrict__ x2,
                              int NL, int owned, int C) {
    long long i = (long long)blockIdx.x * blockDim.x + threadIdx.x;
    long long n = (long long)NL * C;
    if (i >= n) return;
    const int row = (int)(i / C);
    const int c   = (int)(i - (long long)row * C);
    float v = 0.0f;
    if (row < owned) {
        v = agg[i] * dis[row] + b[c];
        v = v > 0.0f ? v : 0.0f;
    }
    x2[i] = v;
}

// ---------------------------------------------------------------------------
// out[i,c] = agg[i,c]*dis[i] + b[c]   for i < owned (final layer, no relu)
// ---------------------------------------------------------------------------
__global__ void final_bias_scale(const float* __restrict__ agg,
                                 const float* __restrict__ dis,
                                 const float* __restrict__ b,
                                 float* __restrict__ out,
                                 int owned, int C) {
    long long i = (long long)blockIdx.x * blockDim.x + threadIdx.x;
    long long n = (long long)owned * C;
    if (i >= n) return;
    const int row = (int)(i / C);
    const int c   = (int)(i - (long long)row * C);
    out[i] = agg[i] * dis[row] + b[c];
}

// ---------------------------------------------------------------------------
// Host-side orchestration (graph-capture safe: launches only, all on stream)
// ---------------------------------------------------------------------------
extern "C" void kernel_launch(void* const* d_in, const int* in_sizes, int n_in,
                              void* d_out, int out_size, void* d_ws, size_t ws_size,
                              hipStream_t stream) {
    const float* x    = (const float*)d_in[0];
    const float* dis  = (const float*)d_in[1];
    const float* w1   = (const float*)d_in[2];
    const float* b1   = (const float*)d_in[3];
    const float* w2   = (const float*)d_in[4];
    const float* b2   = (const float*)d_in[5];
    const int*   erow = (const int*)d_in[6];
    const int*   ecol = (const int*)d_in[7];
    // d_in[8] is num_owned as a device scalar; derive on host instead.

    const int NL    = in_sizes[1];              // 55000
    const int IN_C  = in_sizes[0] / NL;         // 128
    const int HID   = in_sizes[3];              // 128
    const int OUT   = in_sizes[5];              // 64
    const int NE    = in_sizes[6];              // 800000
    const int owned = out_size / OUT;           // 50000

    float* bufH = (float*)d_ws;                       // NL x HID
    float* bufA = bufH + (size_t)NL * HID;            // NL x HID

    const int mTiles = (NL + 15) / 16;
    const int gemmBlocks = (mTiles + 3) / 4;          // 4 waves (m-tiles) / block

    // ---- Layer 1 ----
    {   // h1 = (X @ W1) * dis  -> bufH     (N = 128 -> NT = 8, 64 KB LDS)
        const size_t shmem = (size_t)IN_C * HID * sizeof(float);
        gcn_gemm_scale_wmma<8><<<gemmBlocks, 128, shmem, stream>>>(
            x, w1, dis, bufH, NL, IN_C);
    }
    {   // agg1 = 0 -> bufA
        const long long n = (long long)NL * HID;
        zero_f32<<<(unsigned)((n + 255) / 256), 256, 0, stream>>>(bufA, n);
    }
    {   // scatter: bufA[row] += bufH[col]
        const long long total = (long long)NE * (HID / 4);
        scatter_add_edges<<<(unsigned)((total + 255) / 256), 256, 0, stream>>>(
            bufH, erow, ecol, bufA, NE, HID);
    }
    {   // x2 = pad(relu(agg1*dis + b1)) -> bufH
        const long long n = (long long)NL * HID;
        relu_bias_pad<<<(unsigned)((n + 255) / 256), 256, 0, stream>>>(
            bufA, dis, b1, bufH, NL, owned, HID);
    }

    // ---- Layer 2 ----
    {   // h2 = (x2 @ W2) * dis -> bufA (NL x OUT)   (N = 64 -> NT = 4, 32 KB LDS)
        const size_t shmem = (size_t)HID * OUT * sizeof(float);
        gcn_gemm_scale_wmma<4><<<gemmBlocks, 128, shmem, stream>>>(
            bufH, w2, dis, bufA, NL, HID);
    }
    {   // agg2 = 0 -> bufH (reuse first NL*OUT floats)
        const long long n = (long long)NL * OUT;
        zero_f32<<<(unsigned)((n + 255) / 256), 256, 0, stream>>>(bufH, n);
    }
    {   // scatter: bufH[row] += bufA[col]
        const long long total = (long long)NE * (OUT / 4);
        scatter_add_edges<<<(unsigned)((total + 255) / 256), 256, 0, stream>>>(
            bufA, erow, ecol, bufH, NE, OUT);
    }
    {   // out = agg2[:owned]*dis + b2
        const long long n = (long long)owned * OUT;
        final_bias_scale<<<(unsigned)((n + 255) / 256), 256, 0, stream>>>(
            bufH, dis, b2, (float*)d_out, owned, OUT);
    }
}